// GravNetOp_35003983463182
// MI455X (gfx1250) — compile-verified
//
#include <hip/hip_runtime.h>
#include <hip/hip_bf16.h>
#include <math.h>

typedef __attribute__((ext_vector_type(2))) float v2f;
typedef __attribute__((ext_vector_type(4))) float v4f;
typedef __attribute__((ext_vector_type(8))) float v8f;

#define C_IN   64
#define S_DIM  4
#define P_DIM  64
#define KNN    32
#define C_OUT  128
#define FEAT   (C_IN + 2 * P_DIM)   // 192
#define RT     4                    // row tiles per wave in GEMM kernels

// D = A(16x4 f32) * B(4x16 f32) + C(16x16 f32), one instruction, wave32.
// A: lanes 0-15 -> M=lane, K={0,1}; lanes 16-31 -> K={2,3}.  B symmetric.
// C/D: VGPR r: lanes 0-15 -> M=r, lanes 16-31 -> M=r+8; N = lane&15.
__device__ __forceinline__ v8f wmma_f32_k4(v2f a, v2f b, v8f c) {
  return __builtin_amdgcn_wmma_f32_16x16x4_f32(
      false, a, false, b, (short)0, c, false, false);
}

// ---------------------------------------------------------------------------
// Kernel 1: space = x@W_space + b_space ; prop = x@W_prop + b_prop
// Block = 160 threads (5 waves): waves 0..3 -> prop col tiles, wave 4 -> space.
// Each wave owns RT=4 row tiles so every B fragment feeds 4 WMMAs.
// ---------------------------------------------------------------------------
__global__ void gravnet_linear_kernel(const float* __restrict__ x,
                                      const float* __restrict__ Wsp,
                                      const float* __restrict__ bsp,
                                      const float* __restrict__ Wpr,
                                      const float* __restrict__ bpr,
                                      float* __restrict__ space,
                                      float* __restrict__ prop) {
  const int wave = threadIdx.x >> 5;
  const int lane = threadIdx.x & 31;
  const int sub  = lane & 15;
  const int half = lane >> 4;
  const int rowbase = blockIdx.x * (16 * RT);

  const float* xr[RT];
#pragma unroll
  for (int t = 0; t < RT; ++t)
    xr[t] = x + (size_t)(rowbase + 16 * t + sub) * C_IN + half * 2;

  v8f acc[RT];
#pragma unroll
  for (int t = 0; t < RT; ++t) acc[t] = (v8f){};

  if (wave < 4) {
    const int colbase = wave * 16;
#pragma unroll 4
    for (int k0 = 0; k0 < C_IN; k0 += 4) {
      const int ks = k0 + half * 2;
      v2f b;
      b.x = Wpr[(ks + 0) * P_DIM + colbase + sub];
      b.y = Wpr[(ks + 1) * P_DIM + colbase + sub];
#pragma unroll
      for (int t = 0; t < RT; ++t) {
        v2f a = *(const v2f*)(xr[t] + k0);
        acc[t] = wmma_f32_k4(a, b, acc[t]);
      }
    }
    const float bias = bpr[colbase + sub];
#pragma unroll
    for (int t = 0; t < RT; ++t)
#pragma unroll
      for (int r = 0; r < 8; ++r) {
        const int m = 16 * t + r + half * 8;
        prop[(size_t)(rowbase + m) * P_DIM + colbase + sub] = acc[t][r] + bias;
      }
  } else {
    // space tile: only columns 0..3 are stored; columns 4..15 of D are
    // never read, so clamp the B column index instead of predicating loads
    // (keeps EXEC full -> no per-load saveexec blocks).
    const int subc = (sub < S_DIM) ? sub : (S_DIM - 1);
#pragma unroll 4
    for (int k0 = 0; k0 < C_IN; k0 += 4) {
      const int ks = k0 + half * 2;
      v2f b;
      b.x = Wsp[(ks + 0) * S_DIM + subc];
      b.y = Wsp[(ks + 1) * S_DIM + subc];
#pragma unroll
      for (int t = 0; t < RT; ++t) {
        v2f a = *(const v2f*)(xr[t] + k0);
        acc[t] = wmma_f32_k4(a, b, acc[t]);
      }
    }
    if (sub < S_DIM) {
      const float bias = bsp[sub];
#pragma unroll
      for (int t = 0; t < RT; ++t)
#pragma unroll
        for (int r = 0; r < 8; ++r) {
          const int m = 16 * t + r + half * 8;
          space[(size_t)(rowbase + m) * S_DIM + sub] = acc[t][r] + bias;
        }
    }
  }
}

// ---------------------------------------------------------------------------
// Kernel 2: per-event brute-force kNN (K=32) over S_DIM=4 coordinates.
// One wave per 16 queries; one v_wmma_f32_16x16x4_f32 per candidate 16-tile
// (inner dim is exactly 4). Ranking key s = |c|^2 - 2 q.c ; |q|^2 re-added
// only for the final distances. Lanes 0-15 own query rows: threshold +
// replace-max top-32 list in LDS.
// ---------------------------------------------------------------------------
__global__ void gravnet_knn_kernel(const float* __restrict__ space, int M,
                                   int* __restrict__ nidx,
                                   float* __restrict__ distsq) {
  const int e    = blockIdx.y;
  const int lane = threadIdx.x & 31;
  const int sub  = lane & 15;
  const int half = lane >> 4;
  const int qbase = e * M + blockIdx.x * 16;

  __shared__ float sTile[16][17];
  __shared__ float listD[16][KNN];
  __shared__ int   listI[16][KNN];

  const v2f a  = *(const v2f*)(space + (size_t)(qbase + sub) * S_DIM + half * 2);
  const v4f qv = *(const v4f*)(space + (size_t)(qbase + sub) * S_DIM);
  const float sqQ = qv.x * qv.x + qv.y * qv.y + qv.z * qv.z + qv.w * qv.w;

  int   cnt    = 0;
  float curMax = 3.4e38f;

  for (int c0 = 0; c0 < M; c0 += 16) {
    const int cand = e * M + c0 + sub;   // lanes L and L+16 load the same cand
    const v2f b  = *(const v2f*)(space + (size_t)cand * S_DIM + half * 2);
    const v4f cv = *(const v4f*)(space + (size_t)cand * S_DIM);
    const float sqC = cv.x * cv.x + cv.y * cv.y + cv.z * cv.z + cv.w * cv.w;

    v8f d = {};
    d = wmma_f32_k4(a, b, d);            // d[r] = q_{r+half*8} . c_{sub}

#pragma unroll
    for (int r = 0; r < 8; ++r)
      sTile[r + half * 8][sub] = sqC - 2.0f * d[r];
    __syncthreads();

    if (lane < 16) {
      for (int j = 0; j < 16; ++j) {
        const float s  = sTile[lane][j];
        const int   ci = c0 + j;
        if (cnt < KNN) {
          listD[lane][cnt] = s;
          listI[lane][cnt] = ci;
          if (++cnt == KNN) {
            float m = -3.4e38f;
            for (int t = 0; t < KNN; ++t) m = fmaxf(m, listD[lane][t]);
            curMax = m;
          }
        } else if (s < curMax) {
          int am = 0;
          float mv = listD[lane][0], sv = -3.4e38f;
          for (int t = 1; t < KNN; ++t) {
            const float v = listD[lane][t];
            if (v > mv) { sv = mv; mv = v; am = t; }
            else if (v > sv) sv = v;
          }
          listD[lane][am] = s;
          listI[lane][am] = ci;
          curMax = fmaxf(sv, s);
        }
      }
    }
    __syncthreads();
  }

  if (lane < 16) {
    const int q = qbase + lane;
    for (int t = 0; t < KNN; ++t) {
      nidx[(size_t)q * KNN + t]   = e * M + listI[lane][t];
      distsq[(size_t)q * KNN + t] = fmaxf(sqQ + listD[lane][t], 0.0f);
    }
  }
}

// ---------------------------------------------------------------------------
// Kernel 3: weighted gather -> fmean/fmax, assemble feat = [x | fmean | fmax].
// Block = 256 threads = 4 nodes x 64 channels. Weights exp(-10 d^2) and
// neighbor indices staged once per node in LDS (not 64x per channel).
// ---------------------------------------------------------------------------
__global__ void gravnet_aggregate_kernel(const float* __restrict__ x,
                                         const float* __restrict__ prop,
                                         const int* __restrict__ nidx,
                                         const float* __restrict__ distsq,
                                         float* __restrict__ feat) {
  __shared__ float wsm[4][KNN];
  __shared__ int   ism[4][KNN];

  const int tid     = threadIdx.x;
  const int nodeblk = blockIdx.x * 4;

  if (tid < 4 * KNN) {
    const int ln = tid >> 5;          // KNN == 32
    const int k  = tid & 31;
    const size_t o = (size_t)(nodeblk + ln) * KNN + k;
    wsm[ln][k] = __expf(-10.0f * distsq[o]);
    ism[ln][k] = nidx[o];
  }
  __syncthreads();

  const int ln   = tid >> 6;
  const int ch   = tid & 63;
  const int node = nodeblk + ln;

  float acc = 0.0f, mx = -3.4e38f;
#pragma unroll 4
  for (int k = 0; k < KNN; ++k) {
    const int   j = ism[ln][k];
    const float v = prop[(size_t)j * P_DIM + ch] * wsm[ln][k];
    acc += v;
    mx = fmaxf(mx, v);
  }
  float* f = feat + (size_t)node * FEAT;
  f[ch]                = x[(size_t)node * C_IN + ch];
  f[C_IN + ch]         = acc * (1.0f / KNN);
  f[C_IN + P_DIM + ch] = mx;
}

// ---------------------------------------------------------------------------
// Kernel 4: out = relu(feat @ W_out + b_out), 192 -> 128.
// Block = 256 threads (8 waves); wave w -> col tile w, RT=4 row tiles:
// every B fragment is reused by 4 WMMAs (48 k-steps x 4 = 192 WMMA/wave).
// ---------------------------------------------------------------------------
__global__ void gravnet_out_kernel(const float* __restrict__ feat,
                                   const float* __restrict__ Wout,
                                   const float* __restrict__ bout,
                                   float* __restrict__ out) {
  const int wave = threadIdx.x >> 5;
  const int lane = threadIdx.x & 31;
  const int sub  = lane & 15;
  const int half = lane >> 4;
  const int rowbase = blockIdx.x * (16 * RT);
  const int colbase = wave * 16;

  const float* fr[RT];
#pragma unroll
  for (int t = 0; t < RT; ++t)
    fr[t] = feat + (size_t)(rowbase + 16 * t + sub) * FEAT + half * 2;

  v8f acc[RT];
#pragma unroll
  for (int t = 0; t < RT; ++t) acc[t] = (v8f){};

#pragma unroll 4
  for (int k0 = 0; k0 < FEAT; k0 += 4) {
    const int ks = k0 + half * 2;
    v2f b;
    b.x = Wout[(ks + 0) * C_OUT + colbase + sub];
    b.y = Wout[(ks + 1) * C_OUT + colbase + sub];
#pragma unroll
    for (int t = 0; t < RT; ++t) {
      v2f a = *(const v2f*)(fr[t] + k0);
      acc[t] = wmma_f32_k4(a, b, acc[t]);
    }
  }
  const float bias = bout[colbase + sub];
#pragma unroll
  for (int t = 0; t < RT; ++t)
#pragma unroll
    for (int r = 0; r < 8; ++r) {
      const int m = 16 * t + r + half * 8;
      out[(size_t)(rowbase + m) * C_OUT + colbase + sub] =
          fmaxf(acc[t][r] + bias, 0.0f);
    }
}

// ---------------------------------------------------------------------------
extern "C" void kernel_launch(void* const* d_in, const int* in_sizes, int n_in,
                              void* d_out, int out_size, void* d_ws, size_t ws_size,
                              hipStream_t stream) {
  const float* x    = (const float*)d_in[0];
  // d_in[1] = row_splits (uniform; only sizes needed)
  const float* Wsp  = (const float*)d_in[2];
  const float* bsp  = (const float*)d_in[3];
  const float* Wpr  = (const float*)d_in[4];
  const float* bpr  = (const float*)d_in[5];
  const float* Wout = (const float*)d_in[6];
  const float* bout = (const float*)d_in[7];

  const int N = in_sizes[0] / C_IN;      // 40960
  const int E = in_sizes[1] - 1;         // 10
  const int M = N / E;                   // 4096

  char* ws = (char*)d_ws;
  size_t off = 0;
  auto alloc = [&](size_t bytes) {
    void* p = ws + off;
    off = (off + bytes + 255) & ~(size_t)255;
    return p;
  };
  float* space = (float*)alloc((size_t)N * S_DIM * sizeof(float));
  float* prop  = (float*)alloc((size_t)N * P_DIM * sizeof(float));
  int*   nix   = (int*)  alloc((size_t)N * KNN * sizeof(int));
  float* dist  = (float*)alloc((size_t)N * KNN * sizeof(float));
  float* feat  = (float*)alloc((size_t)N * FEAT * sizeof(float));
  (void)ws_size;

  gravnet_linear_kernel<<<N / (16 * RT), 160, 0, stream>>>(x, Wsp, bsp, Wpr,
                                                           bpr, space, prop);
  gravnet_knn_kernel<<<dim3(M / 16, E), 32, 0, stream>>>(space, M, nix, dist);
  gravnet_aggregate_kernel<<<N / 4, 256, 0, stream>>>(x, prop, nix, dist, feat);
  gravnet_out_kernel<<<N / (16 * RT), 256, 0, stream>>>(feat, Wout, bout,
                                                        (float*)d_out);
}